// GCN_37563783971339
// MI455X (gfx1250) — compile-verified
//
#include <hip/hip_runtime.h>

#define F_IN 32
#define HID  16
#define NCLS 2

typedef __attribute__((ext_vector_type(2))) float v2f;
typedef __attribute__((ext_vector_type(8))) float v8f;

// ---------------------------------------------------------------- degree
__global__ void deg_kernel(const int* __restrict__ dst, float* __restrict__ deg, int E) {
    int e = blockIdx.x * blockDim.x + threadIdx.x;
    if (e < E) atomicAdd(&deg[dst[e]], 1.0f);
}

__global__ void dinv_kernel(const float* __restrict__ deg, float* __restrict__ dinv, int N) {
    int n = blockIdx.x * blockDim.x + threadIdx.x;
    if (n < N) dinv[n] = rsqrtf(deg[n] + 1.0f);
}

// ---------------------------------------------------------------- h1 = x @ W1 via fp32 WMMA
// One wave32 per 16-node tile. K=32 consumed as 8 chained 16x16x4 fp32 WMMAs.
// A (16x4 f32): lanes 0-15 hold M=0..15; VGPR0 = {K0 | K2}, VGPR1 = {K1 | K3} per lane half.
// B (4x16 f32): same striping with N across lanes.
// D (16x16 f32): vgpr r = row r (lanes 0-15) / row r+8 (lanes 16-31), col = lane&15.
__global__ void gemm1_wmma(const float* __restrict__ x, const float* __restrict__ W1,
                           float* __restrict__ h1, int N) {
    const int wavesPerBlock = blockDim.x >> 5;
    const int wave = blockIdx.x * wavesPerBlock + (threadIdx.x >> 5);
    const int lane = threadIdx.x & 31;
    const int nTiles = (N + 15) >> 4;
    if (wave >= nTiles) return;                 // wave-uniform: EXEC stays all-1s for WMMA

    const int m     = lane & 15;
    const int khalf = lane >> 4;                // 0: K{0,1}, 1: K{2,3}
    const int row0  = wave * 16;
    int row = row0 + m;
    int rclamp = row < N ? row : (N - 1);       // clamp loads for a ragged last tile

    v8f acc = {};
    const float* xrow = x + (long)rclamp * F_IN;
#pragma unroll
    for (int kb = 0; kb < F_IN / 4; ++kb) {
        const int k0 = kb * 4 + khalf * 2;
        v2f a, b;
        a.x = xrow[k0 + 0];
        a.y = xrow[k0 + 1];
        b.x = W1[(k0 + 0) * HID + m];
        b.y = W1[(k0 + 1) * HID + m];
        acc = __builtin_amdgcn_wmma_f32_16x16x4_f32(
            /*neg_a=*/false, a, /*neg_b=*/false, b,
            /*c_mod=*/(short)0, acc, /*reuse_a=*/false, /*reuse_b=*/false);
    }

    // D-matrix store: lane-half owns rows r / r+8 of the tile, column = m.
    float* orow = h1 + (long)(row0 + 8 * khalf) * HID + m;
    if (row0 + 16 <= N) {
        // full tile (the only case when 16 | N): unguarded, no exec-mask churn
#pragma unroll
        for (int r = 0; r < 8; ++r) orow[(long)r * HID] = acc[r];
    } else {
        // ragged last tile
#pragma unroll
        for (int r = 0; r < 8; ++r) {
            if (row0 + r + 8 * khalf < N) orow[(long)r * HID] = acc[r];
        }
    }
}

// ---------------------------------------------------------------- edge scatter, 16 channels
// 4 threads per edge, each owns one float4 chunk: coalesced 64B gather per row.
__global__ void edge_scatter16(const int* __restrict__ src, const int* __restrict__ dst,
                               const float* __restrict__ dinv, const float* __restrict__ h,
                               float* __restrict__ agg, int E) {
    long idx = (long)blockIdx.x * blockDim.x + threadIdx.x;
    long e = idx >> 2;
    int  q = (int)(idx & 3);
    if (e >= E) return;
    int s = src[e], d = dst[e];
    float c = dinv[s] * dinv[d];
    float4 v = ((const float4*)h)[(long)s * 4 + q];
    float* ap = agg + (long)d * HID + q * 4;
    atomicAdd(ap + 0, v.x * c);
    atomicAdd(ap + 1, v.y * c);
    atomicAdd(ap + 2, v.z * c);
    atomicAdd(ap + 3, v.w * c);
}

// ---------------------------------------------------------------- layer-1 epilogue + h2 = relu(.)@W2
__global__ void node_l1_l2(const float* __restrict__ agg1, const float* __restrict__ h1,
                           const float* __restrict__ dinv, const float* __restrict__ b1,
                           const float* __restrict__ W2, float* __restrict__ h2, int N) {
    int n = blockIdx.x * blockDim.x + threadIdx.x;
    if (n >= N) return;
    float di = dinv[n];
    float d2 = di * di;
    float o0 = 0.f, o1 = 0.f;
#pragma unroll
    for (int j = 0; j < HID; ++j) {
        float v = agg1[(long)n * HID + j] + h1[(long)n * HID + j] * d2 + b1[j];
        v = v > 0.f ? v : 0.f;                  // relu
        o0 = fmaf(v, W2[j * NCLS + 0], o0);
        o1 = fmaf(v, W2[j * NCLS + 1], o1);
    }
    h2[(long)n * NCLS + 0] = o0;
    h2[(long)n * NCLS + 1] = o1;
}

// ---------------------------------------------------------------- edge scatter, 2 channels
__global__ void edge_scatter2(const int* __restrict__ src, const int* __restrict__ dst,
                              const float* __restrict__ dinv, const float* __restrict__ h2,
                              float* __restrict__ agg2, int E) {
    long e = (long)blockIdx.x * blockDim.x + threadIdx.x;
    if (e >= E) return;
    int s = src[e], d = dst[e];
    float c = dinv[s] * dinv[d];
    float2 v = ((const float2*)h2)[s];
    atomicAdd(&agg2[(long)d * NCLS + 0], v.x * c);
    atomicAdd(&agg2[(long)d * NCLS + 1], v.y * c);
}

// ---------------------------------------------------------------- layer-2 epilogue + graph pooling
__global__ void node_pool(const float* __restrict__ agg2, const float* __restrict__ h2,
                          const float* __restrict__ dinv, const float* __restrict__ b2,
                          const int* __restrict__ batch, float* __restrict__ pooled,
                          float* __restrict__ cnt, int N) {
    int n = blockIdx.x * blockDim.x + threadIdx.x;
    if (n >= N) return;
    float di = dinv[n];
    float d2 = di * di;
    float o0 = agg2[(long)n * NCLS + 0] + h2[(long)n * NCLS + 0] * d2 + b2[0];
    float o1 = agg2[(long)n * NCLS + 1] + h2[(long)n * NCLS + 1] * d2 + b2[1];
    int g = batch[n];
    atomicAdd(&pooled[g * NCLS + 0], o0);
    atomicAdd(&pooled[g * NCLS + 1], o1);
    atomicAdd(&cnt[g], 1.0f);
}

// ---------------------------------------------------------------- mean + log_softmax
__global__ void logsoftmax_kernel(const float* __restrict__ pooled, const float* __restrict__ cnt,
                                  float* __restrict__ out, int G) {
    int g = blockIdx.x * blockDim.x + threadIdx.x;
    if (g >= G) return;
    float c = fmaxf(cnt[g], 1.0f);
    float p0 = pooled[g * NCLS + 0] / c;
    float p1 = pooled[g * NCLS + 1] / c;
    float m  = fmaxf(p0, p1);
    float l  = logf(__expf(p0 - m) + __expf(p1 - m));
    out[g * NCLS + 0] = p0 - m - l;
    out[g * NCLS + 1] = p1 - m - l;
}

extern "C" void kernel_launch(void* const* d_in, const int* in_sizes, int n_in,
                              void* d_out, int out_size, void* d_ws, size_t ws_size,
                              hipStream_t stream) {
    const float* x    = (const float*)d_in[0];
    const float* W1   = (const float*)d_in[1];
    const float* b1   = (const float*)d_in[2];
    const float* W2   = (const float*)d_in[3];
    const float* b2   = (const float*)d_in[4];
    const int*   eidx = (const int*)d_in[5];
    const int*   batch= (const int*)d_in[6];
    float* out = (float*)d_out;

    const int N = in_sizes[6];
    const int E = in_sizes[5] / 2;
    const int G = out_size / NCLS;
    const int* src = eidx;
    const int* dst = eidx + E;

    // workspace layout
    float* deg    = (float*)d_ws;                       // N
    float* dinv   = deg    + N;                         // N
    float* h1     = dinv   + N;                         // N*16
    float* agg1   = h1     + (long)N * HID;             // N*16
    float* h2     = agg1   + (long)N * HID;             // N*2
    float* agg2   = h2     + (long)N * NCLS;            // N*2
    float* pooled = agg2   + (long)N * NCLS;            // G*2
    float* cnt    = pooled + (long)G * NCLS;            // G

    hipMemsetAsync(deg,    0, sizeof(float) * N, stream);
    hipMemsetAsync(agg1,   0, sizeof(float) * (long)N * HID, stream);
    hipMemsetAsync(agg2,   0, sizeof(float) * (long)N * NCLS, stream);
    hipMemsetAsync(pooled, 0, sizeof(float) * (long)G * NCLS, stream);
    hipMemsetAsync(cnt,    0, sizeof(float) * G, stream);

    const int BT = 256;
    // degree + norm
    deg_kernel <<<(E + BT - 1) / BT, BT, 0, stream>>>(dst, deg, E);
    dinv_kernel<<<(N + BT - 1) / BT, BT, 0, stream>>>(deg, dinv, N);

    // layer 1 dense GEMM on the matrix pipe (8 waves / block, 1 tile / wave)
    const int nTiles = (N + 15) / 16;
    gemm1_wmma<<<(nTiles + 7) / 8, BT, 0, stream>>>(x, W1, h1, N);

    // layer 1 message passing (4 threads per edge)
    long t1 = (long)E * 4;
    edge_scatter16<<<(int)((t1 + BT - 1) / BT), BT, 0, stream>>>(src, dst, dinv, h1, agg1, E);

    // layer 1 epilogue + layer 2 dense
    node_l1_l2<<<(N + BT - 1) / BT, BT, 0, stream>>>(agg1, h1, dinv, b1, W2, h2, N);

    // layer 2 message passing
    edge_scatter2<<<(E + BT - 1) / BT, BT, 0, stream>>>(src, dst, dinv, h2, agg2, E);

    // layer 2 epilogue + pooling
    node_pool<<<(N + BT - 1) / BT, BT, 0, stream>>>(agg2, h2, dinv, b2, batch, pooled, cnt, N);

    // mean + log_softmax
    logsoftmax_kernel<<<(G + BT - 1) / BT, BT, 0, stream>>>(pooled, cnt, out, G);
}